// GNNConv_36000415875584
// MI455X (gfx1250) — compile-verified
//
#include <hip/hip_runtime.h>
#include <hip/hip_bf16.h>

typedef float v2f __attribute__((ext_vector_type(2)));
typedef float v4f __attribute__((ext_vector_type(4)));
typedef float v8f __attribute__((ext_vector_type(8)));

// Pointer types matching __builtin_amdgcn_global_load_async_to_lds_b128:
// (int __vector(4) addrspace(1)*, int __vector(4) addrspace(3)*, imm, imm)
typedef int gnn_v4i __attribute__((vector_size(16)));
typedef __attribute__((address_space(1))) gnn_v4i* gnn_gptr4;
typedef __attribute__((address_space(3))) gnn_v4i* gnn_lptr4;

// Problem shape (fixed by the reference): adj is [B=8, N=2048, N=2048] f32.
#define GNN_B 8
#define GNN_N 2048

// CDNA5 async global->LDS path (ASYNCcnt), guarded so absence falls back to
// the proven global_load_b128 + ds_store_b128 staging.
#if defined(__gfx1250__) && __has_builtin(__builtin_amdgcn_global_load_async_to_lds_b128)
#define GNN_ASYNC 1
#endif

#if defined(__gfx1250__) && __has_builtin(__builtin_amdgcn_s_wait_asynccnt)
#define GNN_WAIT_ASYNC(n) __builtin_amdgcn_s_wait_asynccnt(n)
#else
#define GNN_WAIT_ASYNC(n) asm volatile("s_wait_asynccnt %0" ::"i"(n) : "memory")
#endif

// ---------------------------------------------------------------------------
// Kernel 1: deg[b,i] = 1 + sum_j adj[b,i,j];  d[b,i] = deg>0 ? rsqrt(deg) : 0
//
// One wave per 16-row strip. 16x64 f32 tiles staged into LDS (double-buffered
// async-to-LDS copies when available), row sums computed on the matrix pipe:
// V_WMMA_F32_16X16X4_F32 with all-ones B gives D[m,n] = sum_k A[m,k].
// LDS row stride = 68 DWORDs -> fragment reads (dword 4*m + k + {0,2}) touch
// all 64 banks exactly once (conflict-free) and stay 16B/8B aligned.
// ---------------------------------------------------------------------------
__global__ __launch_bounds__(128) void gnn_deg_kernel(
    const float* __restrict__ adj, float* __restrict__ d)
{
    constexpr int N = GNN_N;
    constexpr int TILE_C = 64;       // columns staged per iteration
    constexpr int LDS_STRIDE = 68;   // dwords per LDS row
    constexpr int BUF = 16 * LDS_STRIDE;

    __shared__ float tile[4][2 * BUF];   // 4 waves * 2 buffers * 4352 B = 34816 B

    const int lane = threadIdx.x & 31;
    const int wave = threadIdx.x >> 5;
    const int strip = blockIdx.x * 4 + wave;          // 0 .. B*N/16-1 = 1023
    const int b    = strip >> 7;                      // strip / (N/16)
    const int row0 = (strip & 127) << 4;

    const float* __restrict__ base = adj + ((size_t)b * N + row0) * (size_t)N;
    float* __restrict__ lds = tile[wave];

    const int half = lane >> 4;       // 0: K={0,1}, 1: K={2,3}
    const int mrow = lane & 15;       // M index of this lane's A fragment

    // Per-lane staging coordinates: lane covers chunks t = lane, lane+32, ...
    // (256 float4 chunks per 16x64 tile; r = t>>4, c4 = t&15; coalesced.)
    v8f acc = {0.f, 0.f, 0.f, 0.f, 0.f, 0.f, 0.f, 0.f};
    const v2f ones = {1.0f, 1.0f};

#if defined(GNN_ASYNC)
    // Issue one tile's 8 async b128 copies per lane into buffer `buf`.
    auto issue_tile = [&](int c0, int buf) {
        float* dst = lds + buf * BUF;
        #pragma unroll
        for (int t = lane; t < 256; t += 32) {
            const int r  = t >> 4;
            const int c4 = t & 15;
            __builtin_amdgcn_global_load_async_to_lds_b128(
                (gnn_gptr4)(base + (size_t)r * N + c0 + (c4 << 2)),
                (gnn_lptr4)(dst + r * LDS_STRIDE + (c4 << 2)),
                0, 0);
        }
    };

    issue_tile(0, 0);
    for (int c0 = 0; c0 < N; c0 += TILE_C) {
        const int cur = (c0 / TILE_C) & 1;
        const int nxt = c0 + TILE_C;
        if (nxt < N) {                      // wave-uniform branch (EXEC intact)
            issue_tile(nxt, cur ^ 1);
            GNN_WAIT_ASYNC(8);              // current tile's 8/lane have landed
        } else {
            GNN_WAIT_ASYNC(0);
        }
        const float* arow = lds + cur * BUF + mrow * LDS_STRIDE + (half << 1);
        #pragma unroll
        for (int k = 0; k < TILE_C; k += 4) {
            v2f a = *(const v2f*)(arow + k);          // ds_load_b64
            acc = __builtin_amdgcn_wmma_f32_16x16x4_f32(
                false, a, false, ones, (short)0, acc, false, false);
        }
    }
#else
    for (int c0 = 0; c0 < N; c0 += TILE_C) {
        #pragma unroll
        for (int t = lane; t < 256; t += 32) {
            const int r  = t >> 4;
            const int c4 = t & 15;
            const float4 v =
                *(const float4*)(base + (size_t)r * N + c0 + (c4 << 2));
            *(float4*)(lds + r * LDS_STRIDE + (c4 << 2)) = v;
        }
        asm volatile("s_wait_dscnt 0" ::: "memory");
        const float* arow = lds + mrow * LDS_STRIDE + (half << 1);
        #pragma unroll
        for (int k = 0; k < TILE_C; k += 4) {
            v2f a = *(const v2f*)(arow + k);
            acc = __builtin_amdgcn_wmma_f32_16x16x4_f32(
                false, a, false, ones, (short)0, acc, false, false);
        }
    }
#endif

    // acc[j]: lanes 0-15 hold rowsum(row j), lanes 16-31 hold rowsum(row 8+j),
    // replicated across n. Lanes 0 and 16 commit the 16 values of this strip.
    if ((lane & 15) == 0) {
        float* dp = d + (size_t)b * N + row0 + (half << 3);
        #pragma unroll
        for (int j = 0; j < 8; ++j) {
            const float deg = acc[j] + 1.0f;          // + identity diagonal
            dp[j] = (deg > 0.0f) ? rsqrtf(deg) : 0.0f;
        }
    }
}

// ---------------------------------------------------------------------------
// Kernel 2: out[b,i,j] = d[b,i] * d[b,j] * (adj[b,i,j] + (i==j))
// Pure bandwidth, fully coalesced b128 traffic. adj (134 MB) is resident in
// the 192 MB L2 after K1; NT loads (last use) + NT stores keep the output
// stream from evicting it, so HBM ~ one read + one write of the tensor.
// ---------------------------------------------------------------------------
__global__ __launch_bounds__(256) void gnn_scale_kernel(
    const float* __restrict__ adj, const float* __restrict__ d,
    float* __restrict__ out)
{
    constexpr int N = GNN_N;
    const size_t t = (size_t)blockIdx.x * blockDim.x + threadIdx.x; // float4 id
    const int    j4 = (int)(t & (N / 4 - 1));        // float4 column index
    const size_t bi = t >> 9;                        // b*N + i
    const int    i  = (int)(bi & (N - 1));
    const size_t bN = (bi >> 11) << 11;              // b*N

    const size_t off = (bi << 11) + ((size_t)j4 << 2);
    const v4f  a  = __builtin_nontemporal_load((const v4f*)(adj + off));
    const float di = d[bi];
    const v4f  dj = *(const v4f*)(d + bN + ((size_t)j4 << 2));

    const int jb = j4 << 2;
    v4f r;
    r.x = di * dj.x * (a.x + ((i == jb + 0) ? 1.0f : 0.0f));
    r.y = di * dj.y * (a.y + ((i == jb + 1) ? 1.0f : 0.0f));
    r.z = di * dj.z * (a.z + ((i == jb + 2) ? 1.0f : 0.0f));
    r.w = di * dj.w * (a.w + ((i == jb + 3) ? 1.0f : 0.0f));
    __builtin_nontemporal_store(r, (v4f*)(out + off));
}

extern "C" void kernel_launch(void* const* d_in, const int* in_sizes, int n_in,
                              void* d_out, int out_size, void* d_ws, size_t ws_size,
                              hipStream_t stream) {
    (void)in_sizes; (void)n_in; (void)out_size; (void)ws_size;
    const float* adj = (const float*)d_in[0];
    float*       out = (float*)d_out;
    float*       dvec = (float*)d_ws;                 // [B, N] = 64 KB scratch

    // K1: 1024 strips of 16 rows, 4 waves (strips) per 128-thread block.
    const int strips = GNN_B * (GNN_N / 16);          // 1024
    gnn_deg_kernel<<<strips / 4, 128, 0, stream>>>(adj, dvec);

    // K2: one float4 per thread over the whole [B, N, N] tensor.
    const size_t nvec4 = (size_t)GNN_B * GNN_N * GNN_N / 4;   // 8,388,608
    gnn_scale_kernel<<<(int)(nvec4 / 256), 256, 0, stream>>>(adj, dvec, out);
}